// PseudoImageScatter_23622320128570
// MI455X (gfx1250) — compile-verified
//
#include <hip/hip_runtime.h>
#include <hip/hip_bf16.h>
#include <stdint.h>

// PointPillars pseudo-image scatter for MI455X (gfx1250).
// FLOP-free data movement: zero the 256MB canvas (bandwidth bound, ~11us at
// 23.3 TB/s), then scatter 100k x 64 fp32 features into random (y,x) cells of
// 64 feature planes. CDNA5 paths: async store-from-LDS (ASYNCcnt) with
// immediate IOFFSETs and NT temporal hint for the streaming zero-fill;
// NT stores + wave-uniform scalar coords loads for the scatter.

#define HW_ (1024 * 1024)
#define W_ 1024
#define H_ 1024
#define NFEAT 64

typedef float floatx4 __attribute__((ext_vector_type(4)));
typedef float floatx2 __attribute__((ext_vector_type(2)));
// Builtin parameter type per hipcc diagnostic: GCC-vector int[4].
typedef int v4i_gcc __attribute__((vector_size(4 * sizeof(int))));

#if defined(__gfx1250__)
#define AS_GLOBAL __attribute__((address_space(1)))
#define AS_LDS __attribute__((address_space(3)))
#endif

// ---------------------------------------------------------------------------
// Kernel 1: zero-fill d_out (64M floats).
// Fast path (full block in range, which the launcher guarantees for the 256MB
// canvas): one per-thread base address, then 16 async b128 stores whose
// addresses differ only by the instruction's immediate IOFFSET (i*4096B) ->
// one instruction per 16B chunk, cpol=1 (TH=NT) so zeros stream past L2.
// ---------------------------------------------------------------------------
__global__ void __launch_bounds__(256)
zero_fill_kernel(float* __restrict__ out, unsigned long long n_floats) {
  __shared__ __align__(16) float zsmem[64];
  if (threadIdx.x < 64) zsmem[threadIdx.x] = 0.0f;
  __syncthreads();

  const unsigned long long CPT = 16ull;   // 16B chunks per thread
  const unsigned long long TPB = 256ull;  // threads per block
  const unsigned long long block_chunk0 =
      (unsigned long long)blockIdx.x * (CPT * TPB);
  // Thread's chunk i lives at chunk index block_chunk0 + i*TPB + tid,
  // i.e. byte offset (thread base) + i*4096.
  float* base = out + (block_chunk0 + threadIdx.x) * 4ull;
  const bool full_block = (block_chunk0 + CPT * TPB) * 4ull <= n_floats;

#if defined(__gfx1250__) && \
    __has_builtin(__builtin_amdgcn_global_store_async_from_lds_b128)
  AS_LDS v4i_gcc* lsrc = (AS_LDS v4i_gcc*)(&zsmem[0]);
  if (full_block) {
#define ZSTORE(i)                                                     \
  __builtin_amdgcn_global_store_async_from_lds_b128(                  \
      (AS_GLOBAL v4i_gcc*)base, lsrc, /*ioffset=*/(i) * 4096,         \
      /*cpol: TH=NT*/ 1)
    ZSTORE(0);  ZSTORE(1);  ZSTORE(2);  ZSTORE(3);
    ZSTORE(4);  ZSTORE(5);  ZSTORE(6);  ZSTORE(7);
    ZSTORE(8);  ZSTORE(9);  ZSTORE(10); ZSTORE(11);
    ZSTORE(12); ZSTORE(13); ZSTORE(14); ZSTORE(15);
#undef ZSTORE
  } else {
    for (int i = 0; i < (int)CPT; ++i) {
      unsigned long long idx =
          (block_chunk0 + (unsigned long long)i * TPB + threadIdx.x) * 4ull;
      if (idx + 3ull < n_floats) {
        __builtin_amdgcn_global_store_async_from_lds_b128(
            (AS_GLOBAL v4i_gcc*)(out + idx), lsrc, 0, 1);
      }
    }
  }
#if __has_builtin(__builtin_amdgcn_s_wait_asynccnt)
  __builtin_amdgcn_s_wait_asynccnt(0);
#else
  asm volatile("s_wait_asynccnt 0" ::: "memory");
#endif
#else
  floatx4 z = {0.0f, 0.0f, 0.0f, 0.0f};
  if (full_block) {
#pragma unroll
    for (int i = 0; i < (int)CPT; ++i) {
      __builtin_nontemporal_store(z, (floatx4*)(base + (size_t)i * 1024));
    }
  } else {
    for (int i = 0; i < (int)CPT; ++i) {
      unsigned long long idx =
          (block_chunk0 + (unsigned long long)i * TPB + threadIdx.x) * 4ull;
      if (idx + 3ull < n_floats) {
        __builtin_nontemporal_store(z, (floatx4*)(out + idx));
      }
    }
  }
#endif
}

// ---------------------------------------------------------------------------
// Kernel 2: scatter. One wave32 per pillar: lane l handles features 2l, 2l+1.
// p is wave-uniform (all 32 lanes of a surviving wave share it), so force it
// into an SGPR with readfirstlane -> coords reads become scalar loads and the
// output base becomes SADDR+voffset. Feature row load is a coalesced 256B
// float2-per-lane read. Scattered 4B writes use NT hints (never re-read).
// ---------------------------------------------------------------------------
__global__ void __launch_bounds__(256)
scatter_kernel(const float* __restrict__ pf, const int* __restrict__ coords,
               float* __restrict__ out, int n_pillars) {
  const int lane = threadIdx.x & 31;
  int p = blockIdx.x * (blockDim.x >> 5) + (threadIdx.x >> 5);
  if (p >= n_pillars) return;
  // Wave-uniform by construction; make it SGPR so coords loads go scalar.
  p = __builtin_amdgcn_readfirstlane(p);

  // coords row: [?, y, x] (int32 per harness convention).
  const int x = coords[p * 3 + 2];
  const int y = coords[p * 3 + 1];
  if ((unsigned)x >= (unsigned)W_ || (unsigned)y >= (unsigned)H_) return;
  const size_t flat = (size_t)y * W_ + (size_t)x;

  const floatx2 v = *(const floatx2*)(pf + (size_t)p * NFEAT + (size_t)lane * 2);

  float* base = out + flat;
  __builtin_nontemporal_store(v.x, base + (size_t)(2 * lane) * HW_);
  __builtin_nontemporal_store(v.y, base + (size_t)(2 * lane + 1) * HW_);
}

// ---------------------------------------------------------------------------
extern "C" void kernel_launch(void* const* d_in, const int* in_sizes, int n_in,
                              void* d_out, int out_size, void* d_ws,
                              size_t ws_size, hipStream_t stream) {
  const float* pf = (const float*)d_in[0];   // (NUM_PILLARS, 64) fp32
  const int* coords = (const int*)d_in[1];   // (NUM_PILLARS, 3) int
  float* out = (float*)d_out;                // (1, 64, 1024, 1024) fp32

  const int n_pillars = in_sizes[0] / NFEAT;
  const unsigned long long n_floats = (unsigned long long)out_size;

  // Zero-fill: 16B chunks, 16 per thread, 256 threads/block (8 waves).
  const unsigned long long chunks = (n_floats + 3ull) / 4ull;
  const unsigned long long per_block = 16ull * 256ull;
  const int zblocks = (int)((chunks + per_block - 1ull) / per_block);
  zero_fill_kernel<<<zblocks, 256, 0, stream>>>(out, n_floats);

  // Scatter: 8 pillars (8 waves) per 256-thread block.
  const int sblocks = (n_pillars + 7) / 8;
  scatter_kernel<<<sblocks, 256, 0, stream>>>(pf, coords, out, n_pillars);
}